// OTRecurrentGCN_84593675862588
// MI455X (gfx1250) — compile-verified
//
#include <hip/hip_runtime.h>
#include <hip/hip_bf16.h>
#include <math.h>

#define N_NODES 131072
#define N_EDGES 4194304
#define F_IN    64
#define HIDDEN  128

typedef __attribute__((ext_vector_type(2))) float v2f;
typedef __attribute__((ext_vector_type(8))) float v8f;

// ---------------- graph preprocessing ----------------

__global__ void deg_kernel(const int* __restrict__ ei, const float* __restrict__ w,
                           float* __restrict__ deg, int E) {
    int e = blockIdx.x * blockDim.x + threadIdx.x;
    if (e < E) atomicAdd(&deg[ei[e]], w[e]);   // src row of edge_index
}

__global__ void dinv_kernel(float* __restrict__ deg, int N) {
    int i = blockIdx.x * blockDim.x + threadIdx.x;
    if (i < N) {
        float d = deg[i];
        deg[i] = (d > 0.0f) ? rsqrtf(d) : 0.0f;
    }
}

__global__ void normw_kernel(const int* __restrict__ ei, const float* __restrict__ w,
                             const float* __restrict__ dinv, float* __restrict__ nw, int E) {
    int e = blockIdx.x * blockDim.x + threadIdx.x;
    if (e < E) {
        int s = ei[e];
        int d = ei[E + e];
        nw[e] = -dinv[s] * w[e] * dinv[d];
    }
}

// y[i] = -a[i]   (accumulator init so spmm(scale=2) yields 2*S*v - a)
__global__ void init_neg_kernel(const float* __restrict__ a, float* __restrict__ y, int n) {
    int i = blockIdx.x * blockDim.x + threadIdx.x;
    if (i < n) y[i] = -a[i];
}

// y[dst] += scale * norm_w * v[src]; 16 lanes per edge, float4 per lane (F_IN=64)
__global__ void spmm_kernel(const int* __restrict__ ei, const float* __restrict__ nw,
                            const float* __restrict__ v, float* __restrict__ y,
                            float scale, int E) {
    int tid = blockIdx.x * blockDim.x + threadIdx.x;
    int e = tid >> 4;
    if (e >= E) return;
    int f = (tid & 15) * 4;
    int s = ei[e];
    int d = ei[E + e];
    float w = nw[e] * scale;
    const float4 vv = *(const float4*)(v + (size_t)s * F_IN + f);
    float* yp = y + (size_t)d * F_IN + f;
    atomicAdd(yp + 0, w * vv.x);
    atomicAdd(yp + 1, w * vv.y);
    atomicAdd(yp + 2, w * vv.z);
    atomicAdd(yp + 3, w * vv.w);
}

// ---------------- fused gate GEMM (fp32 WMMA) + epilogue ----------------
// Block: 128 threads (4 waves), 16 rows of the node matrix per block.
// Each wave owns 2 N-tiles of 16 cols, for both live gates (Z = gate0, Ht = gate2).
// acc[g][t] accumulated with V_WMMA_F32_16X16X4_F32 over k=0..3 Cheb terms, Kdim=64.
__global__ __launch_bounds__(128) void gate_kernel(
    const float* __restrict__ tx0, const float* __restrict__ tx1,
    const float* __restrict__ tx2, const float* __restrict__ tx3,
    const float* __restrict__ Wx,    // (3,4,64,128)
    const float* __restrict__ bx,    // (3,128)
    const float* __restrict__ bh,    // (3,128)
    const float* __restrict__ Wlin,  // (128,1)
    const float* __restrict__ blin,  // (1,)
    float* __restrict__ out,         // N
    float* __restrict__ logits)      // N
{
    const int r0      = blockIdx.x * 16;
    const int wave    = threadIdx.x >> 5;   // 0..3
    const int lane    = threadIdx.x & 31;
    const int halfSel = lane >> 4;          // 0: K pair kk..kk+1, 1: kk+2..kk+3
    const int l15     = lane & 15;

    v8f accZ[2] = {};   // gate 0
    v8f accH[2] = {};   // gate 2

    #pragma unroll
    for (int k = 0; k < 4; ++k) {
        const float* T = (k == 0) ? tx0 : (k == 1) ? tx1 : (k == 2) ? tx2 : tx3;
        const float* Wg0 = Wx + (size_t)(0 * 4 + k) * F_IN * HIDDEN;
        const float* Wg2 = Wx + (size_t)(2 * 4 + k) * F_IN * HIDDEN;
        const int arow = r0 + l15;
        for (int kk = 0; kk < F_IN; kk += 4) {
            const int k0 = kk + halfSel * 2;
            // A fragment (16x4 fp32): lane<16 holds K=kk,kk+1; lane>=16 holds kk+2,kk+3
            v2f a;
            a.x = T[(size_t)arow * F_IN + k0];
            a.y = T[(size_t)arow * F_IN + k0 + 1];
            #pragma unroll
            for (int t = 0; t < 2; ++t) {
                const int bcol = (wave * 2 + t) * 16 + l15;
                v2f b0, b2;
                b0.x = Wg0[(size_t)(k0    ) * HIDDEN + bcol];
                b0.y = Wg0[(size_t)(k0 + 1) * HIDDEN + bcol];
                b2.x = Wg2[(size_t)(k0    ) * HIDDEN + bcol];
                b2.y = Wg2[(size_t)(k0 + 1) * HIDDEN + bcol];
                accZ[t] = __builtin_amdgcn_wmma_f32_16x16x4_f32(
                    false, a, false, b0, (short)0, accZ[t], false, false);
                accH[t] = __builtin_amdgcn_wmma_f32_16x16x4_f32(
                    false, a, false, b2, (short)0, accH[t], false, false);
            }
        }
    }

    // Epilogue: bias + sigmoid/tanh + (1-Z)*Ht + relu + row reductions.
    // C/D layout: VGPR i -> row (halfSel*8 + i), column n0 + l15.
    __shared__ float s_sum[16];
    __shared__ float s_dot[16];
    if (threadIdx.x < 16) { s_sum[threadIdx.x] = 0.0f; s_dot[threadIdx.x] = 0.0f; }
    __syncthreads();

    float rsum[8] = {0.f,0.f,0.f,0.f,0.f,0.f,0.f,0.f};
    float rdot[8] = {0.f,0.f,0.f,0.f,0.f,0.f,0.f,0.f};
    #pragma unroll
    for (int t = 0; t < 2; ++t) {
        const int col = (wave * 2 + t) * 16 + l15;
        const float bZ = bx[0 * HIDDEN + col] + bh[0 * HIDDEN + col];
        const float bH = bx[2 * HIDDEN + col] + bh[2 * HIDDEN + col];
        const float wl = Wlin[col];
        #pragma unroll
        for (int i = 0; i < 8; ++i) {
            float z  = 1.0f / (1.0f + expf(-(accZ[t][i] + bZ)));
            float ht = tanhf(accH[t][i] + bH);
            float Hv = (1.0f - z) * ht;
            float h  = Hv > 0.0f ? Hv : 0.0f;
            rsum[i] += h;
            rdot[i] += h * wl;
        }
    }
    #pragma unroll
    for (int i = 0; i < 8; ++i) {
        const int row = halfSel * 8 + i;
        atomicAdd(&s_sum[row], rsum[i]);
        atomicAdd(&s_dot[row], rdot[i]);
    }
    __syncthreads();
    if (threadIdx.x < 16) {
        const int grow = r0 + threadIdx.x;
        const float s = s_sum[threadIdx.x];
        logits[grow] = sqrtf(s);
        out[grow]    = s_dot[threadIdx.x] + blin[0];
    }
}

// ---------------- launch ----------------

extern "C" void kernel_launch(void* const* d_in, const int* in_sizes, int n_in,
                              void* d_out, int out_size, void* d_ws, size_t ws_size,
                              hipStream_t stream) {
    const float* x    = (const float*)d_in[0];
    const int*   ei   = (const int*)d_in[1];      // (2, E)
    const float* ew   = (const float*)d_in[2];
    const float* Wx   = (const float*)d_in[3];    // (3,4,64,128)
    // d_in[4] = W_h : provably unused (H == 0)
    const float* bx   = (const float*)d_in[5];    // (3,128)
    const float* bh   = (const float*)d_in[6];    // (3,128)
    const float* Wlin = (const float*)d_in[7];    // (128,1)
    const float* blin = (const float*)d_in[8];    // (1,)

    float* out    = (float*)d_out;                // first N: h @ W_lin + b
    float* logits = out + N_NODES;                // next  N: sqrt(rowsum(relu(H)))

    char* ws = (char*)d_ws;
    const size_t DEG_OFF = 0;
    const size_t NW_OFF  = DEG_OFF + (size_t)N_NODES * 4;            // 0.5 MB
    const size_t TX1_OFF = NW_OFF  + (size_t)N_EDGES * 4;            // +16 MB
    const size_t TX2_OFF = TX1_OFF + (size_t)N_NODES * F_IN * 4;     // +32 MB
    const size_t TX3_OFF = TX2_OFF + (size_t)N_NODES * F_IN * 4;     // +32 MB
    float* deg = (float*)(ws + DEG_OFF);
    float* nw  = (float*)(ws + NW_OFF);
    float* tx1 = (float*)(ws + TX1_OFF);
    float* tx2 = (float*)(ws + TX2_OFF);
    float* tx3 = (float*)(ws + TX3_OFF);

    const int E = N_EDGES, N = N_NODES, NF = N_NODES * F_IN;

    hipMemsetAsync(deg, 0, (size_t)N * 4, stream);
    hipMemsetAsync(tx1, 0, (size_t)NF * 4, stream);

    deg_kernel  <<<E / 256, 256, 0, stream>>>(ei, ew, deg, E);
    dinv_kernel <<<N / 256, 256, 0, stream>>>(deg, N);
    normw_kernel<<<E / 256, 256, 0, stream>>>(ei, ew, deg, nw, E);

    // Tx1 = S x
    spmm_kernel<<<(E * 16) / 256, 256, 0, stream>>>(ei, nw, x, tx1, 1.0f, E);
    // Tx2 = 2 S Tx1 - x
    init_neg_kernel<<<NF / 256, 256, 0, stream>>>(x, tx2, NF);
    spmm_kernel<<<(E * 16) / 256, 256, 0, stream>>>(ei, nw, tx1, tx2, 2.0f, E);
    // Tx3 = 2 S Tx2 - Tx1
    init_neg_kernel<<<NF / 256, 256, 0, stream>>>(tx1, tx3, NF);
    spmm_kernel<<<(E * 16) / 256, 256, 0, stream>>>(ei, nw, tx2, tx3, 2.0f, E);

    // Fused gate GEMMs + activations + reductions + final linear
    gate_kernel<<<N / 16, 128, 0, stream>>>(x, tx1, tx2, tx3, Wx, bx, bh, Wlin, blin,
                                            out, logits);
}